// HighAggregation_15118284881956
// MI455X (gfx1250) — compile-verified
//
#include <hip/hip_runtime.h>

// Problem constants (match reference)
#define D_FEAT 128

// ---------------------------------------------------------------------------
// Zero the output accumulator (harness poisons d_out with 0xAA).
// ---------------------------------------------------------------------------
__global__ __launch_bounds__(256) void zero_out_kernel(float4* __restrict__ out4, int n4) {
    int i = blockIdx.x * blockDim.x + threadIdx.x;
    if (i < n4) {
        out4[i] = make_float4(0.f, 0.f, 0.f, 0.f);
    }
}

// Wave-uniform broadcast via v_readlane_b32 (wave32). Result is SGPR-uniform,
// so downstream address math goes to the SALU.
__device__ __forceinline__ int   bcast_i(int v, int lane)   { return __builtin_amdgcn_readlane(v, lane); }
__device__ __forceinline__ float bcast_f(float v, int lane) {
    return __int_as_float(__builtin_amdgcn_readlane(__float_as_int(v), lane));
}

// One edge's worth of work: wave reads x[col] row (512 B, 4B128 loads across
// lanes), scales by val, scatter-adds into out[row] with hardware f32 atomics.
__device__ __forceinline__ void process_edge(
    const float* __restrict__ x, float* __restrict__ out,
    int rr, int cc, float vv, int lane)
{
    const float4 xv = *(const float4*)(x + (long long)cc * D_FEAT + (lane << 2));
    float* o = out + (long long)rr * D_FEAT + (lane << 2);
    unsafeAtomicAdd(o + 0, xv.x * vv);
    unsafeAtomicAdd(o + 1, xv.y * vv);
    unsafeAtomicAdd(o + 2, xv.z * vv);
    unsafeAtomicAdd(o + 3, xv.w * vv);
}

// ---------------------------------------------------------------------------
// Edge-parallel SpMM scatter-add.
//   wave (32 lanes) <-> chunk of 32 edges.
//   Phase 1: lane l loads (row, col, val) of edge base+l  -> coalesced,
//            then prefetches x[col] row                   -> global_prefetch_b8.
//   Phase 2: broadcast edge i's metadata via v_readlane (SGPR-uniform),
//            lane l handles features [4l, 4l+4).
// ---------------------------------------------------------------------------
__global__ __launch_bounds__(256) void spmm_scatter_kernel(
    const float* __restrict__ x,     // [N, 128]
    const int*   __restrict__ rows,  // [nnz]
    const int*   __restrict__ cols,  // [nnz]
    const float* __restrict__ vals,  // [nnz]
    float*       __restrict__ out,   // [N, 128]
    int nnz)
{
    const int lane        = threadIdx.x & 31;
    const int waveInBlock = threadIdx.x >> 5;
    const long long waveId = (long long)blockIdx.x * (blockDim.x >> 5) + waveInBlock;
    const long long base   = waveId * 32;
    if (base >= nnz) return;

    const int nEdges = min(32, (int)(nnz - base));

    // Coalesced per-lane edge metadata load.
    int   r = 0, c = 0;
    float v = 0.f;
    const long long myE = base + lane;
    if (myE < nnz) {
        r = rows[myE];
        c = cols[myE];
        v = vals[myE];
        // Prefetch this edge's gather row (gfx1250 global_prefetch_b8).
        __builtin_prefetch(x + (long long)c * D_FEAT, 0, 1);
    }

    if (nEdges == 32) {
        // Fast path: immediate-lane readlane broadcasts, fully unrolled so the
        // scheduler can keep many gathers + atomics in flight.
        #pragma unroll
        for (int i = 0; i < 32; ++i) {
            process_edge(x, out, bcast_i(r, i), bcast_i(c, i), bcast_f(v, i), lane);
        }
    } else {
        // Tail path (last wave only): dynamic-lane readlane.
        for (int i = 0; i < nEdges; ++i) {
            process_edge(x, out, bcast_i(r, i), bcast_i(c, i), bcast_f(v, i), lane);
        }
    }
}

// ---------------------------------------------------------------------------
// Launch: inputs are (x_high f32[N*128], L_rows i32[nnz], L_cols i32[nnz],
// L_vals f32[nnz]); output f32[N*128].
// ---------------------------------------------------------------------------
extern "C" void kernel_launch(void* const* d_in, const int* in_sizes, int n_in,
                              void* d_out, int out_size, void* d_ws, size_t ws_size,
                              hipStream_t stream) {
    const float* x    = (const float*)d_in[0];
    const int*   rows = (const int*)d_in[1];
    const int*   cols = (const int*)d_in[2];
    const float* vals = (const float*)d_in[3];
    float*       out  = (float*)d_out;

    const int nnz = in_sizes[1];

    // 1) zero the accumulator
    {
        const int n4 = out_size / 4;
        const int blocks = (n4 + 255) / 256;
        zero_out_kernel<<<blocks, 256, 0, stream>>>((float4*)out, n4);
    }

    // 2) scatter-add
    {
        const long long waves  = ((long long)nnz + 31) / 32;
        const int wavesPerBlk  = 256 / 32; // 8 waves per 256-thread block (wave32)
        const int blocks       = (int)((waves + wavesPerBlk - 1) / wavesPerBlk);
        spmm_scatter_kernel<<<blocks, 256, 0, stream>>>(x, rows, cols, vals, out, nnz);
    }
}